// HausdorffLoss_48567490183322
// MI455X (gfx1250) — compile-verified
//
#include <hip/hip_runtime.h>

// Problem constants (from reference setup_inputs): output [8,2,512,512] f32, target [8,1,512,512] i32
#define BB 8
#define HH 512
#define WW 512
#define HW (HH*WW)           // 262144 = 2^18
#define NPIX (BB*HW)         // 2097152
#define NMASK 2
#define BIGF  1.0e6f         // reference BIG
#define BIGSQ 1.0e12f        // reference best0 = BIG*BIG
#define ROWS_PER_BLK 4
#define NBLK_RED 2048

typedef __attribute__((ext_vector_type(4))) unsigned int u32x4;
typedef __attribute__((ext_vector_type(8))) unsigned int u32x8;

// ---------------- Stage 0: zero the per-(mask,batch) max slots ----------------
__global__ void hd_init(unsigned* __restrict__ maxbits) {
    if (threadIdx.x < NMASK*BB) maxbits[threadIdx.x] = 0u;
}

// ---------------- Stage 1: vertical 1-D distance, squared ---------------------
// One thread per (mask, b, w) column; adjacent threads -> adjacent w (coalesced).
__global__ void hd_vert(const float* __restrict__ outp,
                        const int*  __restrict__ targ,
                        float* __restrict__ buf /* [2][B][H][W] g^2 */) {
    int t = blockIdx.x * blockDim.x + threadIdx.x;
    if (t >= NMASK*BB*WW) return;
    int maskIdx = t / (BB*WW);
    int rem = t - maskIdx*BB*WW;
    int b = rem / WW;
    int w = rem - b*WW;
    float*       col = buf  + (size_t)maskIdx*NPIX + (size_t)b*HW + w;
    const float* so  = outp + (size_t)b*2*HW + HW + w;   // output channel 1
    const int*   st  = targ + (size_t)b*HW + w;

    // forward scan (matches lax.scan: d = row ? d+1 : 0, init BIG)
    float d = BIGF;
    for (int h = 0; h < HH; ++h) {
        bool m = maskIdx ? (so[(size_t)h*WW] > 0.5f) : (st[(size_t)h*WW] > 0);
        d = m ? d + 1.0f : 0.0f;
        col[(size_t)h*WW] = d;
    }
    // backward scan, combine, square
    d = BIGF;
    for (int h = HH-1; h >= 0; --h) {
        bool m = maskIdx ? (so[(size_t)h*WW] > 0.5f) : (st[(size_t)h*WW] > 0);
        d = m ? d + 1.0f : 0.0f;
        float g = fminf(col[(size_t)h*WW], d);
        col[(size_t)h*WW] = g * g;
    }
}

// ---------------- Stage 2: per-row parabola min (the hot loop) ----------------
// One wave32 per row; each wave DMAs its 2KB g^2 row into LDS with the TDM,
// then every lane computes 16 output columns j = lane + 32*i via
// dt2[j] = min_k (g2[k] + (j-k)^2), reading g2[k] as a b128 LDS broadcast.
// Writes dt2 back in place and atomically maxes the per-(mask,batch) dt2.
__global__ void __launch_bounds__(32*ROWS_PER_BLK)
hd_rowmin(float* __restrict__ buf, unsigned* __restrict__ maxbits) {
    __shared__ float smem[ROWS_PER_BLK * WW];            // 8 KB of the 320 KB WGP LDS
    const int lane = threadIdx.x & 31;
    const int wave = threadIdx.x >> 5;
    const int row  = blockIdx.x * ROWS_PER_BLK + wave;   // 0 .. 2*B*H-1
    const int maskIdx = row / (BB*HH);
    const int rr = row - maskIdx*BB*HH;
    const int b  = rr / HH;
    float* rowPtr = buf + (size_t)maskIdx*NPIX + (size_t)rr*WW;
    float* sm = smem + wave*WW;

    // ---- Tensor Data Mover: global -> LDS, 1 x 512 tile of 4-byte elements ----
    unsigned long long ga = (unsigned long long)(size_t)rowPtr;
    unsigned ldsOff = (unsigned)(size_t)sm;              // LDS byte offset (low 32 bits)
    u32x4 g0;
    g0[0] = 1u;                                          // count=1, user mode, no gather
    g0[1] = ldsOff;                                      // lds_addr
    g0[2] = (unsigned)ga;                                // global_addr[31:0]
    g0[3] = ((unsigned)(ga >> 32) & 0x01FFFFFFu)         // global_addr[56:32]
          | (2u << 30);                                  // type = 2 ("image")
    u32x8 g1;
    g1[0] = (2u << 16);                                  // wg_mask=0, data_size=2 (4B)
    g1[1] = ((unsigned)WW & 0xFFFFu) << 16;              // tensor_dim0[15:0] @ bits 63:48
    g1[2] = (1u << 16);                                  // tensor_dim0 hi=0, tensor_dim1=1
    g1[3] = ((unsigned)WW) << 16;                        // tile_dim0 @ bits 127:112
    g1[4] = 1u;                                          // tile_dim1=1, tile_dim2=0
    g1[5] = (unsigned)WW;                                // tensor_dim0_stride low 32
    g1[6] = 0u;
    g1[7] = 0u;
    asm volatile("tensor_load_to_lds %0, %1" :: "s"(g0), "s"(g1) : "memory");
    asm volatile("s_wait_tensorcnt 0x0" ::: "memory");   // TDM done -> LDS readable

    // ---- min-plus inner loop ----
    // Per k4 step: one ds_load_b128 broadcast feeds 4*16 = 64 independent
    // {fma, min, sub} triplets -> ~192 VALU per DS access, 16 parallel dep chains.
    float best[16], diff[16];
    #pragma unroll
    for (int i = 0; i < 16; ++i) {
        best[i] = BIGSQ;
        diff[i] = (float)(lane + 32*i);                  // j - k at k = 0
    }
    const float4* sm4 = (const float4*)sm;
    for (int k4 = 0; k4 < WW/4; ++k4) {
        float4 g = sm4[k4];                              // same addr all lanes: b128 broadcast
        #pragma unroll
        for (int i = 0; i < 16; ++i) {
            float d0 = diff[i];
            float d1 = d0 - 1.0f;
            float d2 = d0 - 2.0f;
            float d3 = d0 - 3.0f;
            float c0 = fmaf(d0, d0, g.x);
            float c1 = fmaf(d1, d1, g.y);
            float c2 = fmaf(d2, d2, g.z);
            float c3 = fmaf(d3, d3, g.w);
            best[i] = fminf(fminf(fminf(best[i], c0), fminf(c1, c2)), c3);
            diff[i] = d0 - 4.0f;
        }
    }

    float m = 0.0f;
    #pragma unroll
    for (int i = 0; i < 16; ++i) {
        rowPtr[lane + 32*i] = best[i];                   // coalesced 128B per i
        m = fmaxf(m, best[i]);
    }
    // wave32 max-reduce, then one atomic per row (uint bits: valid, all >= 0)
    #pragma unroll
    for (int off = 16; off > 0; off >>= 1)
        m = fmaxf(m, __shfl_xor(m, off, 32));
    if (lane == 0)
        atomicMax(&maxbits[maskIdx*BB + b], __float_as_uint(m));
}

// ---------------- Stage 3: 1/max(dt^2) per (mask,batch) -----------------------
// sqrt-free normalization: dtm^2 = dt2 / max(dt2); max==0 -> divide by 1
__global__ void hd_inv(const unsigned* __restrict__ maxbits, float* __restrict__ inv_mx) {
    int i = threadIdx.x;
    if (i < NMASK*BB) {
        float mx = __uint_as_float(maxbits[i]);
        inv_mx[i] = (mx > 0.0f) ? (1.0f / mx) : 1.0f;
    }
}

// ---------------- Stage 4: fused loss map + per-block partial sums ------------
__global__ void hd_partial(const float* __restrict__ outp,
                           const int*  __restrict__ targ,
                           const float* __restrict__ buf,
                           const float* __restrict__ inv_mx,
                           float* __restrict__ partials) {
    __shared__ float red[256];
    float acc = 0.0f;
    for (int idx = blockIdx.x*256 + threadIdx.x; idx < NPIX; idx += NBLK_RED*256) {
        int b = idx >> 18;                               // HW = 2^18
        int r = idx & (HW - 1);
        float o1 = outp[(size_t)b*2*HW + HW + r];
        float gt = (float)targ[idx];
        float ds = o1 - gt;
        float dtm = buf[(size_t)NPIX + idx] * inv_mx[BB + b]   // seg_dtm^2
                  + buf[idx]               * inv_mx[b];        // gt_dtm^2
        acc += ds * ds * dtm;
    }
    red[threadIdx.x] = acc;
    __syncthreads();
    for (int s = 128; s > 0; s >>= 1) {
        if (threadIdx.x < s) red[threadIdx.x] += red[threadIdx.x + s];
        __syncthreads();
    }
    if (threadIdx.x == 0) partials[blockIdx.x] = red[0];
}

// ---------------- Stage 5: deterministic final reduction ----------------------
__global__ void hd_final(const float* __restrict__ partials, float* __restrict__ out) {
    __shared__ float red[256];
    float acc = 0.0f;
    for (int i = threadIdx.x; i < NBLK_RED; i += 256) acc += partials[i];
    red[threadIdx.x] = acc;
    __syncthreads();
    for (int s = 128; s > 0; s >>= 1) {
        if (threadIdx.x < s) red[threadIdx.x] += red[threadIdx.x + s];
        __syncthreads();
    }
    if (threadIdx.x == 0) out[0] = red[0] * (1.0f / (float)NPIX);
}

extern "C" void kernel_launch(void* const* d_in, const int* in_sizes, int n_in,
                              void* d_out, int out_size, void* d_ws, size_t ws_size,
                              hipStream_t stream) {
    const float* outp = (const float*)d_in[0];   // [8,2,512,512] f32
    const int*   targ = (const int*)d_in[1];     // [8,1,512,512] i32
    float* outv = (float*)d_out;                 // [1] f32

    // workspace layout: buf[2][B][H][W] f32 (16MB) | maxbits[16] u32 | inv_mx[16] f32 | partials[2048] f32
    float*    buf      = (float*)d_ws;
    unsigned* maxbits  = (unsigned*)(buf + 2*(size_t)NPIX);
    float*    inv_mx   = (float*)(maxbits + NMASK*BB);
    float*    partials = inv_mx + NMASK*BB;

    hd_init<<<1, 32, 0, stream>>>(maxbits);
    hd_vert<<<(NMASK*BB*WW + 255)/256, 256, 0, stream>>>(outp, targ, buf);
    hd_rowmin<<<(NMASK*BB*HH)/ROWS_PER_BLK, 32*ROWS_PER_BLK, 0, stream>>>(buf, maxbits);
    hd_inv<<<1, 32, 0, stream>>>(maxbits, inv_mx);
    hd_partial<<<NBLK_RED, 256, 0, stream>>>(outp, targ, buf, inv_mx, partials);
    hd_final<<<1, 256, 0, stream>>>(partials, outv);
}